// THPModel_23227183137529
// MI455X (gfx1250) — compile-verified
//
#include <hip/hip_runtime.h>
#include <math.h>
#include <stdint.h>

typedef float v2f __attribute__((ext_vector_type(2)));
typedef float v8f __attribute__((ext_vector_type(8)));

#define S_TOT 8192
#define KTY   50
#define NDEV  100
#define BLK   128            // events per scan block
#define NBLK  (S_TOT / BLK)  // 64
#define KPAD  64             // padded event-type dim (4 WMMA N-tiles)
#define DECAYF 1.0f

// workspace layout (float offsets)
#define WS_MU      0         // 64
#define WS_ALPHA   64        // 2500  alpha[r][c] row-major
#define WS_ADJ     2624      // 10000 adj[v][w] row-major
#define WS_ALPHAT  12624     // 2500  alphaT[c][r] = alpha[r][c]
#define WS_SUMA    15168     // 64  : alpha row sums  sum_k alpha[e][k]
#define WS_SUMADJ  15232     // 128 : adj  col sums  sum_v adj[v][u]
#define WS_MUSUM   15360
#define WS_LOGSUM  15361

__device__ __forceinline__ float softplusf(float x) {
    return (x > 20.f) ? x : log1pf(expf(x));
}

// CDNA5 async copy: global -> LDS, per-lane addresses, tracked with ASYNCcnt.
__device__ __forceinline__ void async_ld_b32(void* lds_ptr, const void* gptr) {
    uint32_t loff = (uint32_t)(uintptr_t)lds_ptr;            // low 32b = LDS offset
    unsigned long long ga = (unsigned long long)(uintptr_t)gptr;
    asm volatile("global_load_async_to_lds_b32 %0, %1, off"
                 :: "v"(loff), "v"(ga) : "memory");
}

// ---------------------------------------------------------------------------
// Kernel 1: softplus transforms, transpose, row/col sums (deterministic)
// ---------------------------------------------------------------------------
__global__ __launch_bounds__(256)
void prep_kernel(const float* __restrict__ mu_raw,
                 const float* __restrict__ alpha_raw,
                 const float* __restrict__ adj_raw,
                 float* __restrict__ ws) {
    int t = threadIdx.x;
    for (int i = t; i < NDEV * NDEV; i += 256) ws[WS_ADJ + i]   = softplusf(adj_raw[i]);
    for (int i = t; i < KTY * KTY;   i += 256) ws[WS_ALPHA + i] = softplusf(alpha_raw[i]);
    for (int i = t; i < KTY;         i += 256) ws[WS_MU + i]    = softplusf(mu_raw[i]);
    __syncthreads();
    for (int i = t; i < KTY * KTY; i += 256) {             // transpose alpha
        int r = i / KTY, c = i % KTY;
        ws[WS_ALPHAT + c * KTY + r] = ws[WS_ALPHA + i];
    }
    if (t < NDEV) {                                        // col sums: sum_v adj[v][t]
        float s = 0.f;
        for (int v = 0; v < NDEV; ++v) s += ws[WS_ADJ + v * NDEV + t];
        ws[WS_SUMADJ + t] = s;
    }
    if (t < KTY) {                                         // row sums: sum_k alpha[t][k]
        float s = 0.f;
        for (int k = 0; k < KTY; ++k) s += ws[WS_ALPHA + t * KTY + k];
        ws[WS_SUMA + t] = s;
    }
    if (t == 0) {
        float s = 0.f;
        for (int k = 0; k < KTY; ++k) s += ws[WS_MU + k];
        ws[WS_MUSUM] = s;
    }
}

// ---------------------------------------------------------------------------
// Kernel 2: blocked Hawkes scan. Single WG, 8 waves.
//   - inter-block excitation: P(128x64) = AdjRows(128x100) @ R(100x64)
//     via V_WMMA_F32_16X16X4_F32 (exact fp32 path)
//   - next block's event data prefetched with GLOBAL_LOAD_ASYNC_TO_LDS_B32
// ---------------------------------------------------------------------------
__global__ __launch_bounds__(256)
void scan_kernel(const float* __restrict__ times,
                 const int*   __restrict__ events,
                 const int*   __restrict__ devices,
                 float* __restrict__ ws) {
    __shared__ float Rs[NDEV * KPAD];   // 25600 B
    __shared__ float Ps[BLK * KPAD];    // 32768 B
    __shared__ float tb[2][BLK];        // double-buffered block data (async target)
    __shared__ int   ub[2][BLK];
    __shared__ int   eb[2][BLK];
    __shared__ float w_s[BLK];          // e^{t_j - tNew}
    __shared__ float intra[256];
    __shared__ float red[256];

    const int tid  = threadIdx.x;
    const int wv   = tid >> 5;          // wave id 0..7 (wave32)
    const int lane = tid & 31;
    const int laneHalf = lane >> 4;
    const int lrow = lane & 15;

    const float* adjW  = ws + WS_ADJ;
    const float* alpTW = ws + WS_ALPHAT;
    const float* muW   = ws + WS_MU;

    for (int i = tid; i < NDEV * KPAD; i += 256) Rs[i] = 0.f;
    if (tid < BLK) {                    // block 0 loaded directly
        tb[0][tid] = times[tid];
        ub[0][tid] = devices[tid];
        eb[0][tid] = events[tid];
    }
    float tRef    = 0.f;
    float logpart = 0.f;
    __syncthreads();

    for (int blk = 0; blk < NBLK; ++blk) {
        const int base = blk * BLK;
        const int cb = blk & 1, nb = cb ^ 1;
        const float* t_s = tb[cb];
        const int*   u_s = ub[cb];
        const int*   e_s = eb[cb];

        // ---- async prefetch of next block's event data (overlaps all compute)
        if (blk + 1 < NBLK && tid < BLK) {
            const int g = base + BLK + tid;
            async_ld_b32(&tb[nb][tid], times + g);
            async_ld_b32(&ub[nb][tid], devices + g);
            async_ld_b32(&eb[nb][tid], events + g);
        }

        const float tNew = t_s[BLK - 1];
        if (tid < BLK) w_s[tid] = expf(t_s[tid] - tNew);   // <= 1, no overflow
        __syncthreads();

        // ---- P = Adj_I @ R via fp32 WMMA; wave wv owns rows [wv*16, wv*16+16)
        {
            const int i0   = wv * 16;
            const int urow = u_s[i0 + lrow];
            const float* adjRow = adjW + urow * NDEV;

            v8f c[4] = {};
            for (int kk = 0; kk < NDEV / 4; ++kk) {
                const int k0 = kk * 4 + 2 * laneHalf;
                // A (16x4 f32): lanes0-15 -> K=k0,k0+1 ; lanes16-31 -> K=k0+2,k0+3
                v2f a; a.x = adjRow[k0]; a.y = adjRow[k0 + 1];
                const float* rb = &Rs[k0 * KPAD + lrow];
#pragma unroll
                for (int nt = 0; nt < 4; ++nt) {
                    // B (4x16 f32): VGPR r, lane-half h -> row K = r+2h
                    v2f b; b.x = rb[nt * 16]; b.y = rb[KPAD + nt * 16];
                    c[nt] = __builtin_amdgcn_wmma_f32_16x16x4_f32(
                        false, a, false, b, (short)0, c[nt], false, false);
                }
            }
#pragma unroll
            for (int nt = 0; nt < 4; ++nt)
#pragma unroll
                for (int r = 0; r < 8; ++r)
                    Ps[(i0 + r + 8 * laneHalf) * KPAD + nt * 16 + lrow] = c[nt][r];
        }
        __syncthreads();

        // ---- intra-block triangle, 2-way split over j parity (exp-free body)
        float dotv = 0.f;
        {
            const int i  = (tid < BLK) ? tid : tid - BLK;
            const int ei = e_s[i], ui = u_s[i];
            const float* at = alpTW + ei * KTY;   // at[k] = alpha[k][ei], contiguous
            const float* ar = adjW + ui * NDEV;
            float s = 0.f;
            const int j0 = (tid < BLK) ? 0 : 1;
            for (int j = j0; j < i; j += 2)
                s += ar[u_s[j]] * at[e_s[j]] * w_s[j];
            intra[tid] = s;
            if (tid < BLK)
                for (int b = 0; b < KTY; ++b)
                    dotv += Ps[i * KPAD + b] * at[b];
        }
        __syncthreads();

        // ---- intensities + log-likelihood; rescale R (disjoint data, same phase)
        if (tid < BLK) {
            const float ti = t_s[tid];
            const int   ei = e_s[tid];
            const float excit = expf(-DECAYF * (ti - tRef)) * dotv +
                                expf(-DECAYF * (ti - tNew)) * (intra[tid] + intra[tid + BLK]);
            logpart += logf(muW[ei] + excit + 1e-6f);
        }
        const float dec = expf(-DECAYF * (tNew - tRef));
        for (int i = tid; i < NDEV * KPAD; i += 256) Rs[i] *= dec;
        __syncthreads();

        // ---- deterministic parallel scatter: thread owns cells (cell&255)==tid
        for (int j = 0; j < BLK; ++j) {
            const int cell = u_s[j] * KPAD + e_s[j];     // uniform LDS broadcast reads
            if ((cell & 255) == tid) Rs[cell] += w_s[j];
        }
        tRef = tNew;
        __builtin_amdgcn_s_wait_asynccnt(0);             // next block's LDS data ready
        __syncthreads();
    }

    // ---- deterministic tree reduction of log-likelihood
    red[tid] = logpart;
    __syncthreads();
    for (int s = 128; s > 0; s >>= 1) {
        if (tid < s) red[tid] += red[tid + s];
        __syncthreads();
    }
    if (tid == 0) ws[WS_LOGSUM] = red[0];
}

// ---------------------------------------------------------------------------
// Kernel 3: compensator terms + final loss
// ---------------------------------------------------------------------------
__global__ __launch_bounds__(256)
void finalize_kernel(const float* __restrict__ times,
                     const int*   __restrict__ events,
                     const int*   __restrict__ devices,
                     const float* __restrict__ ws,
                     float* __restrict__ out) {
    __shared__ float red[256];
    const int tid = threadIdx.x;
    const float Tend = times[S_TOT - 1];
    float part = 0.f;
    for (int j = tid; j < S_TOT; j += 256) {
        const float tint = (1.f - expf(-DECAYF * (Tend - times[j]))) / DECAYF;
        part += ws[WS_SUMADJ + devices[j]] * ws[WS_SUMA + events[j]] * tint;
    }
    red[tid] = part;
    __syncthreads();
    for (int s = 128; s > 0; s >>= 1) {
        if (tid < s) red[tid] += red[tid + s];
        __syncthreads();
    }
    if (tid == 0) {
        const float term1 = ws[WS_MUSUM] * Tend;
        out[0] = -(ws[WS_LOGSUM] - (term1 + red[0]));
    }
}

// ---------------------------------------------------------------------------
extern "C" void kernel_launch(void* const* d_in, const int* in_sizes, int n_in,
                              void* d_out, int out_size, void* d_ws, size_t ws_size,
                              hipStream_t stream) {
    const float* times     = (const float*)d_in[0];
    const int*   events    = (const int*)  d_in[1];
    const int*   devices   = (const int*)  d_in[2];
    const float* mu_raw    = (const float*)d_in[3];
    const float* alpha_raw = (const float*)d_in[4];
    const float* adj_raw   = (const float*)d_in[5];
    float* ws  = (float*)d_ws;
    float* out = (float*)d_out;

    prep_kernel    <<<1, 256, 0, stream>>>(mu_raw, alpha_raw, adj_raw, ws);
    scan_kernel    <<<1, 256, 0, stream>>>(times, events, devices, ws);
    finalize_kernel<<<1, 256, 0, stream>>>(times, events, devices, ws, out);
}